// GraphNetwork_43258910605908
// MI455X (gfx1250) — compile-verified
//
#include <hip/hip_runtime.h>
#include <math.h>

#define F_IN 64
#define HH   64
#define RT   3
#define NCOL 832   // 128 fs | 64 xskip | 3*64 xr | 3*128 fm | 64 xp
#define GG   64
#define CC   10

typedef float v2f __attribute__((ext_vector_type(2)));
typedef float v8f __attribute__((ext_vector_type(8)));

#if defined(__HIP_DEVICE_COMPILE__)
#if !__has_builtin(__builtin_amdgcn_wmma_f32_16x16x4_f32)
#error "missing __builtin_amdgcn_wmma_f32_16x16x4_f32 on this toolchain"
#endif
#endif

// ---- monotonic float<->uint keys for atomicMax on possibly-negative floats
__device__ __forceinline__ unsigned fkey(float f) {
  unsigned u = __float_as_uint(f);
  return (u & 0x80000000u) ? ~u : (u | 0x80000000u);
}
__device__ __forceinline__ float funkey(unsigned k) {
  unsigned u = (k & 0x80000000u) ? (k & 0x7fffffffu) : ~k;
  return __uint_as_float(u);
}
#define KEY_NEG_INF 0x007fffffu  // fkey(-inf)

// ------------------------------------------------- pack WcatT [NCOL][F_IN]
// Transposed so a B fragment's two K values per lane are contiguous (b64).
__global__ void k_pack(const float* __restrict__ skip_film_W,
                       const float* __restrict__ skip_W,
                       const float* __restrict__ film_lin_W,
                       const float* __restrict__ film_film_W,
                       const float* __restrict__ skip_film_b,
                       const float* __restrict__ film_film_b,
                       const float* __restrict__ gat_W,
                       float* __restrict__ WcatT, float* __restrict__ bias) {
  int idx = blockIdx.x * blockDim.x + threadIdx.x;
  if (idx >= F_IN * NCOL) return;
  int d = idx / NCOL, c = idx % NCOL;
  float w, b = 0.f;
  if (c < 128)      { w = skip_film_W[d * 128 + c];                     b = skip_film_b[c]; }
  else if (c < 192) { w = skip_W[d * 64 + (c - 128)]; }
  else if (c < 384) { int r = (c - 192) >> 6, h = (c - 192) & 63;
                      w = film_lin_W[(r * 64 + d) * 64 + h]; }
  else if (c < 768) { int r = (c - 384) >> 7, j = (c - 384) & 127;
                      w = film_film_W[(r * 64 + d) * 128 + j];          b = film_film_b[r * 128 + j]; }
  else              { w = gat_W[d * 64 + (c - 768)]; }
  WcatT[c * F_IN + d] = w;
  if (d == 0) bias[c] = b;
}

// ------------------------------------------- WMMA f32 GEMM  Y = X @ Wcat
// Block = 64 rows (4 row-tiles) staged in LDS; 8 waves split the 52 col
// tiles; one B fragment (global b64) feeds 4 WMMAs (A frags via ds b64).
// A(16x4 f32): lane L -> M = L&15, VGPRv holds K = v + 2*(L>=16).
// C/D(16x16 f32): VGPR v -> M = v + 8*(lane>=16), N = lane&15.
__global__ void __launch_bounds__(256)
k_gemm(const float* __restrict__ X, const float* __restrict__ WcatT,
       const float* __restrict__ bias, float* __restrict__ Y, int nrows) {
  __shared__ float lds_x[64 * F_IN];                       // 16 KB
  const int row0 = blockIdx.x * 64;

  for (int i = threadIdx.x; i < 64 * (F_IN / 4); i += 256) {
    int r = i >> 4, c4 = i & 15;
    int gr = row0 + r;
    if (gr >= nrows) gr = nrows - 1;                       // clamp tail block
    *reinterpret_cast<float4*>(&lds_x[r * F_IN + c4 * 4]) =
        *reinterpret_cast<const float4*>(X + (size_t)gr * F_IN + c4 * 4);
  }
  __syncthreads();

  const int lane = threadIdx.x & 31;
  const int wave = threadIdx.x >> 5;
  const int lrow = lane & 15;
  const int half = lane >> 4;

  for (int ct = wave; ct < NCOL / 16; ct += 8) {
    const int c0 = ct * 16;
    v8f acc0 = {0.f, 0.f, 0.f, 0.f, 0.f, 0.f, 0.f, 0.f};
    v8f acc1 = acc0, acc2 = acc0, acc3 = acc0;
    const float* bp = WcatT + (size_t)(c0 + lrow) * F_IN + 2 * half;
    const float* ap = &lds_x[lrow * F_IN + 2 * half];
#pragma unroll
    for (int k4 = 0; k4 < 16; ++k4) {
      v2f b  = *reinterpret_cast<const v2f*>(bp + 4 * k4);
      v2f a0 = *reinterpret_cast<const v2f*>(ap + 4 * k4);
      v2f a1 = *reinterpret_cast<const v2f*>(ap + 4 * k4 + 16 * F_IN);
      v2f a2 = *reinterpret_cast<const v2f*>(ap + 4 * k4 + 32 * F_IN);
      v2f a3 = *reinterpret_cast<const v2f*>(ap + 4 * k4 + 48 * F_IN);
      acc0 = __builtin_amdgcn_wmma_f32_16x16x4_f32(false, a0, false, b, (short)0, acc0, false, false);
      acc1 = __builtin_amdgcn_wmma_f32_16x16x4_f32(false, a1, false, b, (short)0, acc1, false, false);
      acc2 = __builtin_amdgcn_wmma_f32_16x16x4_f32(false, a2, false, b, (short)0, acc2, false, false);
      acc3 = __builtin_amdgcn_wmma_f32_16x16x4_f32(false, a3, false, b, (short)0, acc3, false, false);
    }
    const float bv = bias[c0 + lrow];
#pragma unroll
    for (int v = 0; v < 8; ++v) {
      int rbase = row0 + v + 8 * half;
      int col = c0 + lrow;
      if (rbase < nrows)      Y[(size_t)rbase * NCOL + col]        = acc0[v] + bv;
      if (rbase + 16 < nrows) Y[(size_t)(rbase + 16) * NCOL + col] = acc1[v] + bv;
      if (rbase + 32 < nrows) Y[(size_t)(rbase + 32) * NCOL + col] = acc2[v] + bv;
      if (rbase + 48 < nrows) Y[(size_t)(rbase + 48) * NCOL + col] = acc3[v] + bv;
    }
  }
}

// ------------------------------------------ node-local: skip term, GAT dots
__global__ void __launch_bounds__(256)
k_node(const float* __restrict__ Y, const float* __restrict__ att_src,
       const float* __restrict__ att_dst, float* __restrict__ skip_out,
       float* __restrict__ es, float* __restrict__ ed, int n) {
  int node = blockIdx.x * 8 + (threadIdx.x >> 5);
  if (node >= n) return;
  int lane = threadIdx.x & 31;
  const float* y = Y + (size_t)node * NCOL;
  int h0 = 2 * lane;
  v2f beta  = *reinterpret_cast<const v2f*>(y + h0);
  v2f gamma = *reinterpret_cast<const v2f*>(y + 64 + h0);
  v2f xs    = *reinterpret_cast<const v2f*>(y + 128 + h0);
  v2f xp    = *reinterpret_cast<const v2f*>(y + 768 + h0);
  float ps = 0.f, pd = 0.f;
#pragma unroll
  for (int i = 0; i < 2; ++i) {
    float v = gamma[i] * xs[i] + beta[i];
    skip_out[node * 64 + h0 + i] = v > 0.f ? v : 0.f;
    ps += xp[i] * att_src[h0 + i];
    pd += xp[i] * att_dst[h0 + i];
  }
#pragma unroll
  for (int m = 16; m >= 1; m >>= 1) {
    ps += __shfl_xor(ps, m, 32);
    pd += __shfl_xor(pd, m, 32);
  }
  if (lane == 0) { es[node] = ps; ed[node] = pd; }
}

// ------------------------------------------------------------- accum init
__global__ void k_init(float* film_agg, float* gat_agg, int* cnt,
                       unsigned* m_key, float* denom_a, int n) {
  int idx = blockIdx.x * blockDim.x + threadIdx.x;
  if (idx < n * 64) { film_agg[idx] = 0.f; gat_agg[idx] = 0.f; }
  if (idx < n * RT) cnt[idx] = 0;
  if (idx < n)      { m_key[idx] = KEY_NEG_INF; denom_a[idx] = 0.f; }
}

// ------------------------------------------------------------ FiLM edges
__global__ void k_count(const int* __restrict__ ei, const int* __restrict__ et,
                        int* cnt, int E) {
  int e = blockIdx.x * blockDim.x + threadIdx.x;
  if (e >= E) return;
  atomicAdd(&cnt[ei[E + e] * RT + et[e]], 1);
}

__global__ void __launch_bounds__(256)
k_film_edge(const float* __restrict__ Y, const int* __restrict__ ei,
            const int* __restrict__ et, const int* __restrict__ cnt,
            float* __restrict__ film_agg, int E) {
  int e = blockIdx.x * 8 + (threadIdx.x >> 5);
  if (e >= E) return;
  int lane = threadIdx.x & 31;
  int src = ei[e], dst = ei[E + e], r = et[e];
  int c = cnt[dst * RT + r];
  float inv = 1.f / (float)(c > 1 ? c : 1);
  int h0 = 2 * lane;
  const float* ys = Y + (size_t)src * NCOL + 192 + 64 * r;   // xr[r][src]
  const float* yd = Y + (size_t)dst * NCOL + 384 + 128 * r;  // fm[r][dst]
  v2f xr    = *reinterpret_cast<const v2f*>(ys + h0);
  v2f beta  = *reinterpret_cast<const v2f*>(yd + h0);
  v2f gamma = *reinterpret_cast<const v2f*>(yd + 64 + h0);
#pragma unroll
  for (int i = 0; i < 2; ++i) {
    float m = gamma[i] * xr[i] + beta[i];
    m = m > 0.f ? m : 0.f;
    atomicAdd(&film_agg[dst * 64 + h0 + i], m * inv);
  }
}

// -------------------------------------------------------------- GAT edges
__device__ __forceinline__ void gat_edge(const int* ei, int E, int idx,
                                         int& src, int& dst) {
  if (idx < E) { src = ei[idx]; dst = ei[E + idx]; }
  else         { src = dst = idx - E; }                      // self loop
}

__global__ void k_gat_max(const int* __restrict__ ei, const float* __restrict__ es,
                          const float* __restrict__ ed, unsigned* m_key,
                          int E, int n) {
  int idx = blockIdx.x * blockDim.x + threadIdx.x;
  if (idx >= E + n) return;
  int src, dst; gat_edge(ei, E, idx, src, dst);
  float e = es[src] + ed[dst];
  e = e > 0.f ? e : 0.2f * e;                                // leaky_relu 0.2
  atomicMax(&m_key[dst], fkey(e));
}

__global__ void k_gat_denom(const int* __restrict__ ei, const float* __restrict__ es,
                            const float* __restrict__ ed,
                            const unsigned* __restrict__ m_key,
                            float* denom_a, int E, int n) {
  int idx = blockIdx.x * blockDim.x + threadIdx.x;
  if (idx >= E + n) return;
  int src, dst; gat_edge(ei, E, idx, src, dst);
  float e = es[src] + ed[dst];
  e = e > 0.f ? e : 0.2f * e;
  atomicAdd(&denom_a[dst], expf(e - funkey(m_key[dst])));
}

__global__ void __launch_bounds__(256)
k_gat_agg(const float* __restrict__ Y, const int* __restrict__ ei,
          const float* __restrict__ es, const float* __restrict__ ed,
          const unsigned* __restrict__ m_key, const float* __restrict__ denom_a,
          float* __restrict__ gat_agg, int E, int n) {
  int idx = blockIdx.x * 8 + (threadIdx.x >> 5);
  if (idx >= E + n) return;
  int lane = threadIdx.x & 31;
  int src, dst; gat_edge(ei, E, idx, src, dst);
  float e = es[src] + ed[dst];
  e = e > 0.f ? e : 0.2f * e;
  float w = expf(e - funkey(m_key[dst])) / denom_a[dst];
  int h0 = 2 * lane;
  v2f xp = *reinterpret_cast<const v2f*>(Y + (size_t)src * NCOL + 768 + h0);
  atomicAdd(&gat_agg[dst * 64 + h0],     w * xp[0]);
  atomicAdd(&gat_agg[dst * 64 + h0 + 1], w * xp[1]);
}

// --------------------------------------------------------------- features
__global__ void k_feats(const float* __restrict__ skip_out,
                        const float* __restrict__ film_agg,
                        const float* __restrict__ gat_agg,
                        const float* __restrict__ gat_b,
                        float* __restrict__ feats, int n) {
  int idx = blockIdx.x * blockDim.x + threadIdx.x;
  if (idx >= n * 128) return;
  int node = idx >> 7, h2 = idx & 127;
  float v;
  if (h2 < 64) v = skip_out[node * 64 + h2] + film_agg[node * 64 + h2];
  else         v = gat_agg[node * 64 + (h2 - 64)] + gat_b[h2 - 64];
  feats[idx] = v > 0.f ? v : 0.f;
}

// ------------------------------ deterministic segment pool (gb is sorted)
__global__ void __launch_bounds__(128)
k_pool(const float* __restrict__ feats, const int* __restrict__ gb,
       float* __restrict__ pooled, int n) {
  int g = blockIdx.x, h2 = threadIdx.x;
  int lo = 0, hi = n;
  while (lo < hi) { int mid = (lo + hi) >> 1; if (gb[mid] <  g) lo = mid + 1; else hi = mid; }
  int start = lo;
  hi = n;
  while (lo < hi) { int mid = (lo + hi) >> 1; if (gb[mid] <= g) lo = mid + 1; else hi = mid; }
  int end = lo;
  float mx = 0.f, sum = 0.f;                 // feats >= 0 so max-from-0 is exact
  for (int node = start; node < end; ++node) {
    float v = feats[(size_t)node * 128 + h2];
    mx = v > mx ? v : mx;
    sum += v;
  }
  int c = end - start;
  pooled[g * 256 + h2]       = mx;           // cnt==0 -> 0, matches reference
  pooled[g * 256 + 128 + h2] = sum / (float)(c > 0 ? c : 1);
}

// -------------------------------------------------- MLP head + log_softmax
__global__ void __launch_bounds__(256)
k_head(const float* __restrict__ pooled, const float* __restrict__ lin_W,
       const float* __restrict__ lin_b, const float* __restrict__ fc_W,
       const float* __restrict__ fc_b, float* __restrict__ out) {
  __shared__ float hidden[GG][HH];
  __shared__ float logits[GG * CC];
  int t = threadIdx.x;
  for (int s = t; s < GG * HH; s += 256) {
    int i = s >> 6, j = s & 63;
    float acc = lin_b[j];
    for (int k = 0; k < 256; ++k)
      acc += pooled[i * 256 + k] * lin_W[k * HH + j];
    hidden[i][j] = acc > 0.f ? acc : 0.f;
  }
  __syncthreads();
  for (int s = t; s < GG * CC; s += 256) {
    int i = s / CC, c = s % CC;
    float acc = fc_b[c];
    for (int j = 0; j < HH; ++j)
      acc += hidden[i][j] * fc_W[j * CC + c];
    logits[s] = acc;
  }
  __syncthreads();
  if (t < GG) {
    float m = logits[t * CC];
    for (int c = 1; c < CC; ++c) m = fmaxf(m, logits[t * CC + c]);
    float ssum = 0.f;
    for (int c = 0; c < CC; ++c) ssum += expf(logits[t * CC + c] - m);
    float lse = m + logf(ssum);
    for (int c = 0; c < CC; ++c) out[t * CC + c] = logits[t * CC + c] - lse;
  }
}

// -------------------------------------------------------------------- host
extern "C" void kernel_launch(void* const* d_in, const int* in_sizes, int n_in,
                              void* d_out, int out_size, void* d_ws, size_t ws_size,
                              hipStream_t stream) {
  (void)n_in; (void)out_size; (void)ws_size;
  const float* x            = (const float*)d_in[0];
  const int*   ei           = (const int*)  d_in[1];
  const int*   et           = (const int*)  d_in[2];
  const int*   gb           = (const int*)  d_in[3];
  const float* film_lin_W   = (const float*)d_in[4];
  const float* film_film_W  = (const float*)d_in[5];
  const float* film_film_b  = (const float*)d_in[6];
  const float* skip_W       = (const float*)d_in[7];
  const float* skip_film_W  = (const float*)d_in[8];
  const float* skip_film_b  = (const float*)d_in[9];
  const float* gat_W        = (const float*)d_in[10];
  const float* att_src      = (const float*)d_in[11];
  const float* att_dst      = (const float*)d_in[12];
  const float* gat_b        = (const float*)d_in[13];
  const float* lin_W        = (const float*)d_in[14];
  const float* lin_b        = (const float*)d_in[15];
  const float* fc_W         = (const float*)d_in[16];
  const float* fc_b         = (const float*)d_in[17];

  const int n = in_sizes[0] / F_IN;      // 50000
  const int E = in_sizes[2];             // 800000

  char* p = (char*)d_ws;
  auto carve = [&](size_t bytes) -> void* {
    void* q = (void*)p;
    p += (bytes + 255) & ~(size_t)255;
    return q;
  };
  float*    Y        = (float*)   carve((size_t)n * NCOL * 4);
  float*    WcatT    = (float*)   carve((size_t)F_IN * NCOL * 4);
  float*    bias     = (float*)   carve(NCOL * 4);
  float*    skip_out = (float*)   carve((size_t)n * 64 * 4);
  float*    film_agg = (float*)   carve((size_t)n * 64 * 4);
  float*    gat_agg  = (float*)   carve((size_t)n * 64 * 4);
  float*    es       = (float*)   carve((size_t)n * 4);
  float*    ed       = (float*)   carve((size_t)n * 4);
  float*    denom_a  = (float*)   carve((size_t)n * 4);
  unsigned* m_key    = (unsigned*)carve((size_t)n * 4);
  int*      cnt      = (int*)     carve((size_t)n * RT * 4);
  float*    feats    = (float*)   carve((size_t)n * 128 * 4);
  float*    pooled   = (float*)   carve((size_t)GG * 256 * 4);

  k_pack<<<(F_IN * NCOL + 255) / 256, 256, 0, stream>>>(
      skip_film_W, skip_W, film_lin_W, film_film_W, skip_film_b, film_film_b,
      gat_W, WcatT, bias);
  k_init<<<(n * 64 + 255) / 256, 256, 0, stream>>>(film_agg, gat_agg, cnt,
                                                   m_key, denom_a, n);
  k_gemm<<<(n + 63) / 64, 256, 0, stream>>>(x, WcatT, bias, Y, n);
  k_node<<<(n + 7) / 8, 256, 0, stream>>>(Y, att_src, att_dst, skip_out, es, ed, n);
  k_count<<<(E + 255) / 256, 256, 0, stream>>>(ei, et, cnt, E);
  k_film_edge<<<(E + 7) / 8, 256, 0, stream>>>(Y, ei, et, cnt, film_agg, E);
  k_gat_max<<<(E + n + 255) / 256, 256, 0, stream>>>(ei, es, ed, m_key, E, n);
  k_gat_denom<<<(E + n + 255) / 256, 256, 0, stream>>>(ei, es, ed, m_key, denom_a, E, n);
  k_gat_agg<<<(E + n + 7) / 8, 256, 0, stream>>>(Y, ei, es, ed, m_key, denom_a,
                                                 gat_agg, E, n);
  k_feats<<<(n * 128 + 255) / 256, 256, 0, stream>>>(skip_out, film_agg, gat_agg,
                                                     gat_b, feats, n);
  k_pool<<<GG, 128, 0, stream>>>(feats, gb, pooled, n);
  k_head<<<1, 256, 0, stream>>>(pooled, lin_W, lin_b, fc_W, fc_b, (float*)d_out);
}